// TripletLoss_17575006175627
// MI455X (gfx1250) — compile-verified
//
#include <hip/hip_runtime.h>
#include <math.h>
#include <stdint.h>

typedef __attribute__((ext_vector_type(2))) float v2f;
typedef __attribute__((ext_vector_type(8))) float v8f;

#define N_SAMP 4096
#define D_FEAT 2048
#define NB     32          // N_SAMP / BM
#define BM     128
#define BN     128
#define BK     64
#define NKT    (D_FEAT / BK)   // 32 k-stages
#define LDA    68          // BK + 4 pad: 16B-aligned rows, conflict-free b64 frag loads

__device__ __forceinline__ float f_inf()  { return __int_as_float(0x7f800000); }
__device__ __forceinline__ float f_ninf() { return __int_as_float(0xff800000); }

// low 32 bits of a generic pointer into __shared__ == wave-relative LDS byte offset
__device__ __forceinline__ unsigned lds_off(const void* p) {
    return (unsigned)(uintptr_t)p;
}

constexpr float MARGIN = 0.3f;
constexpr float BETA   = 0.4f;

// ---------------------------------------------------------------------------
// Kernel 1: squared row norms + init mining buffers
// ---------------------------------------------------------------------------
__global__ void sqnorm_init_kernel(const float* __restrict__ X,
                                   float* __restrict__ sq,
                                   float* __restrict__ dist_ap,
                                   float* __restrict__ dist_an) {
    __shared__ float red[8];
    const int row = blockIdx.x;
    const float* xr = X + (size_t)row * D_FEAT;
    float s = 0.f;
    for (int k = threadIdx.x; k < D_FEAT; k += 256) {
        float v = xr[k];
        s += v * v;
    }
    #pragma unroll
    for (int m = 16; m >= 1; m >>= 1) s += __shfl_xor(s, m, 32);
    if ((threadIdx.x & 31) == 0) red[threadIdx.x >> 5] = s;
    __syncthreads();
    if (threadIdx.x == 0) {
        float t = 0.f;
        #pragma unroll
        for (int w = 0; w < 8; ++w) t += red[w];
        sq[row]      = t;
        dist_ap[row] = f_ninf();
        dist_an[row] = f_inf();
    }
}

// ---------------------------------------------------------------------------
// Kernel 2: fused Gram-tile WMMA GEMM + distance + masked hard mining.
// Double-buffered LDS tiles filled by async global->LDS copies (ASYNCcnt),
// lower-triangular tile grid, dual-direction mining from each tile.
// ---------------------------------------------------------------------------
__global__ __launch_bounds__(256)
void gram_mine_kernel(const float* __restrict__ X,
                      const long long* __restrict__ targets,
                      const float* __restrict__ sq,
                      float* __restrict__ dist_ap,
                      float* __restrict__ dist_an) {
    __shared__ float As[2][BM * LDA];
    __shared__ float Bs[2][BN * LDA];

    // triangular decode of block index -> (bx, by), by <= bx
    int t  = blockIdx.x;
    int bx = (int)((sqrtf(8.f * (float)t + 1.f) - 1.f) * 0.5f);
    while ((bx + 1) * (bx + 2) / 2 <= t) ++bx;
    while (bx * (bx + 1) / 2 > t) --bx;
    int by = t - bx * (bx + 1) / 2;

    const int rowBase = bx * BM;   // A-tile rows
    const int colBase = by * BN;   // B-tile rows (= Gram columns)

    const int tid  = threadIdx.x;
    const int lane = tid & 31;
    const int wid  = tid >> 5;       // 0..7
    const int wm   = wid & 3;        // 4 waves along M -> 32 rows each
    const int wn   = wid >> 2;       // 2 waves along N -> 64 cols each
    const int ml   = lane & 15;
    const int half = lane >> 4;

    v8f acc[2][4];
    const v8f zero = {0.f, 0.f, 0.f, 0.f, 0.f, 0.f, 0.f, 0.f};
    #pragma unroll
    for (int i = 0; i < 2; ++i)
        #pragma unroll
        for (int j = 0; j < 4; ++j) acc[i][j] = zero;

    // --- async tile staging: each thread issues 8+8 b128 copies per stage ---
    // chunk id = u*256 + tid in [0,2048): row = id/16, col segment = (id%16)*4
    auto issue_stage = [&](int kt, int buf) {
        const int k0 = kt * BK;
        const float* sb = X + k0;          // uniform -> SGPR pair (GVS saddr)
        #pragma unroll
        for (int u = 0; u < 8; ++u) {
            const int id  = u * 256 + tid;
            const int row = id >> 4;
            const int cs  = (id & 15) * 4;
            const unsigned la = lds_off(&As[buf][row * LDA + cs]);
            const unsigned va = (unsigned)(((rowBase + row) * D_FEAT + cs) * 4);
            asm volatile("global_load_async_to_lds_b128 %0, %1, %2 offset:0"
                         :: "v"(la), "v"(va), "s"(sb) : "memory");
            const unsigned lb = lds_off(&Bs[buf][row * LDA + cs]);
            const unsigned vb = (unsigned)(((colBase + row) * D_FEAT + cs) * 4);
            asm volatile("global_load_async_to_lds_b128 %0, %1, %2 offset:0"
                         :: "v"(lb), "v"(vb), "s"(sb) : "memory");
        }
    };

    issue_stage(0, 0);
    if (NKT > 1) issue_stage(1, 1);

    for (int kt = 0; kt < NKT; ++kt) {
        const int buf = kt & 1;
        // retire the group feeding this stage; keep the newer group in flight
        if (kt + 1 < NKT) asm volatile("s_wait_asynccnt 0x10" ::: "memory");
        else              asm volatile("s_wait_asynccnt 0x0"  ::: "memory");
        __syncthreads();   // all waves' async data visible in LDS

        #pragma unroll
        for (int kk = 0; kk < BK / 4; ++kk) {
            const int kcol = kk * 4 + half * 2;
            v2f a[2], b[4];
            #pragma unroll
            for (int i = 0; i < 2; ++i)
                a[i] = *(const v2f*)&As[buf][(wm * 32 + i * 16 + ml) * LDA + kcol];
            #pragma unroll
            for (int j = 0; j < 4; ++j)
                b[j] = *(const v2f*)&Bs[buf][(wn * 64 + j * 16 + ml) * LDA + kcol];
            #pragma unroll
            for (int i = 0; i < 2; ++i)
                #pragma unroll
                for (int j = 0; j < 4; ++j)
                    acc[i][j] = __builtin_amdgcn_wmma_f32_16x16x4_f32(
                        false, a[i], false, b[j], (short)0, acc[i][j],
                        false, false);
        }

        __syncthreads();   // all waves done reading buf before it is refilled
        if (kt + 2 < NKT) issue_stage(kt + 2, buf);
    }

    // ---- epilogue: distances + masked hard mining ----
    // Row direction: r in this block's A rows, c across wave's 64 columns.
    #pragma unroll
    for (int i = 0; i < 2; ++i) {
        #pragma unroll
        for (int v = 0; v < 8; ++v) {
            const int r = rowBase + wm * 32 + i * 16 + half * 8 + v;
            const long long tr = targets[r];
            const float sqr = sq[r];
            float pmax = f_ninf();
            float nmin = f_inf();
            #pragma unroll
            for (int j = 0; j < 4; ++j) {
                const int c = colBase + wn * 64 + j * 16 + ml;
                const float g = acc[i][j][v];
                const float d2 = sqr + sq[c] - 2.f * g;
                const float dist = sqrtf(fmaxf(d2, 1e-12f));
                const bool same = (targets[c] == tr);
                pmax = fmaxf(pmax, same ? dist : f_ninf());
                nmin = fminf(nmin, same ? f_inf() : dist);
            }
            #pragma unroll
            for (int m = 1; m < 16; m <<= 1) {
                pmax = fmaxf(pmax, __shfl_xor(pmax, m, 32));
                nmin = fminf(nmin, __shfl_xor(nmin, m, 32));
            }
            if (ml == 0) {
                atomicMax((int*)(dist_ap + r), __float_as_int(pmax));
                atomicMin((int*)(dist_an + r), __float_as_int(nmin));
            }
        }
    }

    // Column (transposed) direction: treat Gram columns as rows.
    #pragma unroll
    for (int j = 0; j < 4; ++j) {
        const int c = colBase + wn * 64 + j * 16 + ml;
        const long long tc = targets[c];
        const float sqc = sq[c];
        float pmax = f_ninf();
        float nmin = f_inf();
        #pragma unroll
        for (int i = 0; i < 2; ++i) {
            #pragma unroll
            for (int v = 0; v < 8; ++v) {
                const int r = rowBase + wm * 32 + i * 16 + half * 8 + v;
                const float g = acc[i][j][v];
                const float d2 = sq[r] + sqc - 2.f * g;
                const float dist = sqrtf(fmaxf(d2, 1e-12f));
                const bool same = (targets[r] == tc);
                pmax = fmaxf(pmax, same ? dist : f_ninf());
                nmin = fminf(nmin, same ? f_inf() : dist);
            }
        }
        pmax = fmaxf(pmax, __shfl_xor(pmax, 16, 32));
        nmin = fminf(nmin, __shfl_xor(nmin, 16, 32));
        if (half == 0) {
            atomicMax((int*)(dist_ap + c), __float_as_int(pmax));
            atomicMin((int*)(dist_an + c), __float_as_int(nmin));
        }
    }
}

// ---------------------------------------------------------------------------
// Kernel 3: final triplet-penalty reduction -> scalar
// ---------------------------------------------------------------------------
__global__ void finalize_kernel(const float* __restrict__ dist_ap,
                                const float* __restrict__ dist_an,
                                float* __restrict__ out) {
    __shared__ float red[8];
    const float pm = (1.f - BETA) * MARGIN / (1.f + BETA);
    float s = 0.f;
    for (int i = threadIdx.x; i < N_SAMP; i += 256) {
        const float ap = dist_ap[i];
        const float an = dist_an[i];
        s += fmaxf(0.f, -((1.f - BETA) * an - (1.f + BETA) * ap) + pm);
    }
    #pragma unroll
    for (int m = 16; m >= 1; m >>= 1) s += __shfl_xor(s, m, 32);
    if ((threadIdx.x & 31) == 0) red[threadIdx.x >> 5] = s;
    __syncthreads();
    if (threadIdx.x == 0) {
        float t = 0.f;
        #pragma unroll
        for (int w = 0; w < 8; ++w) t += red[w];
        const float loss = t / (float)N_SAMP;
        out[0] = fmaxf(loss, 0.f);
    }
}

// ---------------------------------------------------------------------------
extern "C" void kernel_launch(void* const* d_in, const int* in_sizes, int n_in,
                              void* d_out, int out_size, void* d_ws, size_t ws_size,
                              hipStream_t stream) {
    (void)in_sizes; (void)n_in; (void)out_size; (void)ws_size;
    const float*     X       = (const float*)d_in[0];
    const long long* targets = (const long long*)d_in[1];
    float* out = (float*)d_out;

    float* sq      = (float*)d_ws;
    float* dist_ap = sq + N_SAMP;
    float* dist_an = sq + 2 * N_SAMP;

    sqnorm_init_kernel<<<N_SAMP, 256, 0, stream>>>(X, sq, dist_ap, dist_an);

    const int nblocks = NB * (NB + 1) / 2;   // lower-triangular tile set
    gram_mine_kernel<<<nblocks, 256, 0, stream>>>(X, targets, sq, dist_ap, dist_an);

    finalize_kernel<<<1, 256, 0, stream>>>(dist_ap, dist_an, out);
}